// LoRALayer_24215025615346
// MI455X (gfx1250) — compile-verified
//
#include <hip/hip_runtime.h>
#include <hip/hip_bf16.h>

// CDNA5 / gfx1250 fused LoRA GEMM, double-buffered with async global->LDS:
//   out[m][n] = sum_k x[m][k]*W[n][k]  +  2 * sum_r T[m][r]*A[r][n]
//   with T[m][r] = sum_k x[m][k]*B[k][r]   (rank 16, fused in-pass)
// HBM-bound (~25us floor at 23.3 TB/s): fp32 WMMA keeps reference precision
// for free; async-to-LDS removes the VGPR round trip on the streaming path.

typedef __attribute__((ext_vector_type(2))) float v2f;
typedef __attribute__((ext_vector_type(8))) float v8f;
typedef __attribute__((ext_vector_type(4))) int   i4v;
typedef __attribute__((ext_vector_type(2))) int   i2v;

#define DIM   4096
#define RANK  16
#define KBLK  32
#define XS    36   // LDS row stride (floats): 144B = 16B-aligned, conflict-free frag reads
#define TS    20   // LDS row stride for T tile: conflict-free A-frag reads

#if defined(__has_builtin)
#  if __has_builtin(__builtin_amdgcn_global_load_async_to_lds_b128) && \
      __has_builtin(__builtin_amdgcn_global_load_async_to_lds_b64)
#    define HAVE_ASYNC 1
#  else
#    define HAVE_ASYNC 0
#  endif
#else
#  define HAVE_ASYNC 0
#endif

#if HAVE_ASYNC
typedef __attribute__((address_space(1))) i4v* gp128_t;
typedef __attribute__((address_space(3))) i4v* lp128_t;
typedef __attribute__((address_space(1))) i2v* gp64_t;
typedef __attribute__((address_space(3))) i2v* lp64_t;
#  define ASYNC_B128(g, l) __builtin_amdgcn_global_load_async_to_lds_b128( \
        (gp128_t)(void*)(g), (lp128_t)(void*)(l), 0, 0)
#  define ASYNC_B64(g, l)  __builtin_amdgcn_global_load_async_to_lds_b64( \
        (gp64_t)(void*)(g), (lp64_t)(void*)(l), 0, 0)
#  if __has_builtin(__builtin_amdgcn_s_wait_asynccnt)
#    define WAIT_ASYNC(n) __builtin_amdgcn_s_wait_asynccnt(n)
#  else
#    define WAIT_ASYNC(n) asm volatile("s_wait_asynccnt %0" :: "i"(n) : "memory")
#  endif
#else
#  define ASYNC_B128(g, l) (*(float4*)(l) = *(const float4*)(g))
#  define ASYNC_B64(g, l)  (*(float2*)(l) = *(const float2*)(g))
#  define WAIT_ASYNC(n) do { } while (0)
#endif

__global__ __launch_bounds__(256, 2)
void lora_fused_gemm(const float* __restrict__ x,
                     const float* __restrict__ w,
                     const float* __restrict__ lA,
                     const float* __restrict__ lB,
                     float* __restrict__ out) {
    __shared__ __align__(16) float xs [2][128 * XS];
    __shared__ __align__(16) float wsh[2][128 * XS];
    __shared__ __align__(16) float bs [2][KBLK * RANK];
    __shared__ __align__(16) float ts [128 * TS];

    const int tid   = threadIdx.x;
    const int lane  = tid & 31;
    const int wid   = tid >> 5;     // 0..7
    const int waveM = wid & 1;      // 2 waves along M  -> 64 rows each
    const int waveN = wid >> 1;     // 4 waves along N  -> 32 cols each
    const int l16   = lane & 15;
    const int lhalf = lane >> 4;    // upper half-wave: K+2/K+3 in A/B frags

    const int bm = blockIdx.x * 128;   // M tile origin (M = 16384)
    const int bn = blockIdx.y * 128;   // N tile origin (N = 4096)

    v8f c[4][2];
    v8f t[4];
#pragma unroll
    for (int i = 0; i < 4; ++i) {
        t[i] = (v8f)0.0f;
#pragma unroll
        for (int j = 0; j < 2; ++j) c[i][j] = (v8f)0.0f;
    }

    // Per-thread async transfer set: 4 x-b128 + 4 w-b128 + 1 loraB-b64 = 9 ops
    // (uniform across all waves, so WAIT_ASYNC immediates are wave-uniform).
    auto issue_tiles = [&](int k0, int buf) {
#pragma unroll
        for (int j = 0; j < 4; ++j) {
            const int idx = tid + j * 256;        // 1024 float4 per 128x32 tile
            const int row = idx >> 3;             // 8 float4 per row
            const int col = (idx & 7) << 2;
            ASYNC_B128(x + (size_t)(bm + row) * DIM + k0 + col, &xs[buf][row * XS + col]);
            ASYNC_B128(w + (size_t)(bn + row) * DIM + k0 + col, &wsh[buf][row * XS + col]);
        }
        const int br = tid >> 3;                  // 32x16 loraB tile: 256 float2
        const int bc = (tid & 7) << 1;
        ASYNC_B64(lB + (size_t)(k0 + br) * RANK + bc, &bs[buf][br * RANK + bc]);
    };

    issue_tiles(0, 0);   // prologue: stream first tile set into buffer 0

    for (int k0 = 0, it = 0; k0 < DIM; k0 += KBLK, ++it) {
        const int buf = it & 1;
        __syncthreads();                 // prev compute done; safe to overwrite buf^1
        if (k0 + KBLK < DIM) {
            issue_tiles(k0 + KBLK, buf ^ 1);
            WAIT_ASYNC(9);               // fence current buffer's 9 ops (issued last iter)
        } else {
            WAIT_ASYNC(0);
        }
        __syncthreads();                 // all waves' current-buffer loads visible

#pragma unroll
        for (int kk = 0; kk < KBLK; kk += 4) {
            const int kb = kk + lhalf * 2;
            v2f a[4];
#pragma unroll
            for (int cm = 0; cm < 4; ++cm) {
                const int row = waveM * 64 + cm * 16 + l16;
                a[cm].x = xs[buf][row * XS + kb];
                a[cm].y = xs[buf][row * XS + kb + 1];
            }
            v2f b[2];
#pragma unroll
            for (int cn = 0; cn < 2; ++cn) {
                const int nn = waveN * 32 + cn * 16 + l16;
                b[cn].x = wsh[buf][nn * XS + kb];
                b[cn].y = wsh[buf][nn * XS + kb + 1];
            }
            v2f bb;
            bb.x = bs[buf][kb * RANK + l16];
            bb.y = bs[buf][(kb + 1) * RANK + l16];

#pragma unroll
            for (int cm = 0; cm < 4; ++cm) {
#pragma unroll
                for (int cn = 0; cn < 2; ++cn)
                    c[cm][cn] = __builtin_amdgcn_wmma_f32_16x16x4_f32(
                        false, a[cm], false, b[cn], (short)0, c[cm][cn], false, false);
                t[cm] = __builtin_amdgcn_wmma_f32_16x16x4_f32(
                    false, a[cm], false, bb, (short)0, t[cm], false, false);
            }
        }
    }

    // --- convert T (C-layout) -> A-layout via LDS, scaled by 2.0 ---
    __syncthreads();
    if (waveN == 0) {   // waves 0/1 cover all 128 rows; single writer per element
#pragma unroll
        for (int cm = 0; cm < 4; ++cm) {
#pragma unroll
            for (int j = 0; j < 8; ++j) {
                const int m = waveM * 64 + cm * 16 + j + lhalf * 8;
                ts[m * TS + l16] = 2.0f * t[cm][j];
            }
        }
    }
    __syncthreads();

    // --- C += T(Mx16) @ lora_A(16xN), K = RANK = 16 -> 4 wmma steps ---
#pragma unroll
    for (int kk = 0; kk < RANK; kk += 4) {
        const int kb = kk + lhalf * 2;
        v2f bA[2];
#pragma unroll
        for (int cn = 0; cn < 2; ++cn) {
            const int nn = bn + waveN * 32 + cn * 16 + l16;
            bA[cn].x = lA[(size_t)kb * DIM + nn];
            bA[cn].y = lA[(size_t)(kb + 1) * DIM + nn];
        }
#pragma unroll
        for (int cm = 0; cm < 4; ++cm) {
            v2f aT;
            const int row = waveM * 64 + cm * 16 + l16;
            aT.x = ts[row * TS + kb];
            aT.y = ts[row * TS + kb + 1];
#pragma unroll
            for (int cn = 0; cn < 2; ++cn)
                c[cm][cn] = __builtin_amdgcn_wmma_f32_16x16x4_f32(
                    false, aT, false, bA[cn], (short)0, c[cm][cn], false, false);
        }
    }

    // --- store C (coalesced 64B per half-wave row) ---
#pragma unroll
    for (int cm = 0; cm < 4; ++cm) {
#pragma unroll
        for (int cn = 0; cn < 2; ++cn) {
            const int nn = bn + waveN * 32 + cn * 16 + l16;
#pragma unroll
            for (int j = 0; j < 8; ++j) {
                const int m = bm + waveM * 64 + cm * 16 + j + lhalf * 8;
                out[(size_t)m * DIM + nn] = c[cm][cn][j];
            }
        }
    }
}

extern "C" void kernel_launch(void* const* d_in, const int* in_sizes, int n_in,
                              void* d_out, int out_size, void* d_ws, size_t ws_size,
                              hipStream_t stream) {
    const float* x  = (const float*)d_in[0];   // (4,4096,4096) fp32
    const float* w  = (const float*)d_in[1];   // (4096,4096)
    const float* lA = (const float*)d_in[2];   // (16,4096)
    const float* lB = (const float*)d_in[3];   // (4096,16)
    float* out = (float*)d_out;

    const int M = in_sizes[0] / DIM;           // 16384
    dim3 grid(M / 128, DIM / 128);             // (128, 32)
    lora_fused_gemm<<<grid, 256, 0, stream>>>(x, w, lA, lB, out);
    (void)d_ws; (void)ws_size; (void)n_in; (void)out_size;
}